// ODE_conv_29944511988320
// MI455X (gfx1250) — compile-verified
//
#include <hip/hip_runtime.h>
#include <hip/hip_bf16.h>
#include <math.h>

// ---------------- problem constants ----------------
#define B_SZ    64
#define C_INCH  3
#define S_SZ    16
#define N_OSC   256
#define NUM_CN  32
#define T_STEPS 40
#define DT_STEP 0.125f          // 5.0 / 40
#define FEAT    8192            // 32 * 256
#define KDIM    8192
#define NDIM    8192

typedef float v2f __attribute__((ext_vector_type(2)));
typedef float v8f __attribute__((ext_vector_type(8)));
typedef int   v4i_g __attribute__((vector_size(16)));   // matches builtin's V4i

// ---------------- CDNA5 async global->LDS helpers ----------------
#if defined(__has_builtin)
#  if __has_builtin(__builtin_amdgcn_global_load_async_to_lds_b128)
#    define HAVE_ASYNC_LDS 1
#  endif
#endif
#ifndef HAVE_ASYNC_LDS
#  define HAVE_ASYNC_LDS 0
#endif

__device__ __forceinline__ void async_wait_all() {
#if defined(__has_builtin) && __has_builtin(__builtin_amdgcn_s_wait_asynccnt)
    __builtin_amdgcn_s_wait_asynccnt(0);
#else
    asm volatile("s_wait_asynccnt 0x0" ::: "memory");
#endif
}

#if HAVE_ASYNC_LDS
// 16B global -> LDS async copy (GLOBAL_LOAD_ASYNC_TO_LDS_B128, ASYNCcnt).
// Builtin signature (from clang diagnostic): (v4i AS1*, v4i AS3*, imm, imm).
// Generic->AS casts go through uintptr_t; low 32 bits of a generic LDS
// pointer are the LDS byte offset per the ISA flat-addressing rules.
__device__ __forceinline__ void async_copy16(const float* gsrc, float* ldst) {
    __builtin_amdgcn_global_load_async_to_lds_b128(
        (__attribute__((address_space(1))) v4i_g*)(uintptr_t)gsrc,
        (__attribute__((address_space(3))) v4i_g*)(uintptr_t)ldst,
        0, 0);
}
#endif

// =====================================================================
// Kernel A: fused conv0(3->16,3x3,tanh) + conv1(16->32,3x3,tanh)
// One block per batch image; all operands staged in LDS.
// =====================================================================
__global__ __launch_bounds__(256)
void conv_fused_kernel(const float* __restrict__ x,
                       const float* __restrict__ w0, const float* __restrict__ b0,
                       const float* __restrict__ w1, const float* __restrict__ b1,
                       float* __restrict__ feat) {
    __shared__ float s_w0[16 * 3 * 9];      // 432
    __shared__ float s_b0[16];
    __shared__ float s_w1[32 * 16 * 9];     // 4608
    __shared__ float s_b1[32];
    __shared__ float s_x[C_INCH * 256];
    __shared__ float s_h0[16 * 256];

    const int t = threadIdx.x;
    const int b = blockIdx.x;

    for (int i = t; i < 432;  i += 256) s_w0[i] = w0[i];
    for (int i = t; i < 16;   i += 256) s_b0[i] = b0[i];
    for (int i = t; i < 4608; i += 256) s_w1[i] = w1[i];
    for (int i = t; i < 32;   i += 256) s_b1[i] = b1[i];
    for (int i = t; i < C_INCH * 256; i += 256)
        s_x[i] = x[(size_t)b * C_INCH * 256 + i];
    __syncthreads();

    const int r  = t >> 4;
    const int cc = t & 15;

    // conv0 + tanh -> LDS
    for (int oc = 0; oc < 16; ++oc) {
        float acc = s_b0[oc];
        for (int ic = 0; ic < 3; ++ic) {
            #pragma unroll
            for (int dy = 0; dy < 3; ++dy) {
                int y = r + dy - 1;
                if ((unsigned)y < 16u) {
                    #pragma unroll
                    for (int dx = 0; dx < 3; ++dx) {
                        int xx = cc + dx - 1;
                        if ((unsigned)xx < 16u)
                            acc += s_x[ic * 256 + y * 16 + xx] *
                                   s_w0[((oc * 3 + ic) * 3 + dy) * 3 + dx];
                    }
                }
            }
        }
        s_h0[oc * 256 + t] = tanhf(acc);
    }
    __syncthreads();

    // conv1 + tanh -> feat[b, oc*256 + pixel]
    for (int oc = 0; oc < 32; ++oc) {
        float acc = s_b1[oc];
        for (int ic = 0; ic < 16; ++ic) {
            #pragma unroll
            for (int dy = 0; dy < 3; ++dy) {
                int y = r + dy - 1;
                if ((unsigned)y < 16u) {
                    #pragma unroll
                    for (int dx = 0; dx < 3; ++dx) {
                        int xx = cc + dx - 1;
                        if ((unsigned)xx < 16u)
                            acc += s_h0[ic * 256 + y * 16 + xx] *
                                   s_w1[((oc * 16 + ic) * 3 + dy) * 3 + dx];
                    }
                }
            }
        }
        feat[(size_t)b * FEAT + oc * 256 + t] = tanhf(acc);
    }
}

// =====================================================================
// Kernel B: couplings = feat[64x8192] @ lin_w[8192x8192] + bias
// f32 WMMA 16x16x4, double-buffered LDS tiles fed by async
// global->LDS copies (ASYNCcnt) so chunk k+1's HBM traffic overlaps
// chunk k's WMMAs. WG = 256 threads (8 waves); each wave owns a
// 64(M) x 16(N) strip -> 4 v8f accumulators; 128 N-cols per WG.
// =====================================================================
#define KC 32
#define NT 128
__global__ __launch_bounds__(256)
void gemm_wmma_kernel(const float* __restrict__ A,    // [64, 8192]
                      const float* __restrict__ W,    // [8192, 8192]
                      const float* __restrict__ bias, // [8192]
                      float* __restrict__ C) {        // [64, 8192]
    __shared__ float sA[2][64 * KC];   // 2 x 8 KB
    __shared__ float sB[2][KC * NT];   // 2 x 16 KB

    const int t    = threadIdx.x;
    const int lane = t & 31;
    const int wave = t >> 5;
    const int nBase = blockIdx.x * NT;
    const int koff  = (lane >> 4) * 2;          // f32 frag: K split across lane halves
    const int nLoc  = wave * 16 + (lane & 15);  // this wave's output column (local)

    // per-thread staging coordinates
    int aRow[2], aCol[2], bRow[4], bCol[4];
    #pragma unroll
    for (int j = 0; j < 2; ++j) {
        int idx4 = t + j * 256;
        aRow[j] = idx4 >> 3;           // 8 float4 per 64xKC row
        aCol[j] = (idx4 & 7) * 4;
    }
    #pragma unroll
    for (int j = 0; j < 4; ++j) {
        int idx4 = t + j * 256;
        bRow[j] = idx4 >> 5;           // 32 float4 per KCxNT row
        bCol[j] = (idx4 & 31) * 4;
    }

    auto stage = [&](int kc, int buf) {
#if HAVE_ASYNC_LDS
        #pragma unroll
        for (int j = 0; j < 2; ++j)
            async_copy16(&A[(size_t)aRow[j] * KDIM + kc + aCol[j]],
                         &sA[buf][aRow[j] * KC + aCol[j]]);
        #pragma unroll
        for (int j = 0; j < 4; ++j)
            async_copy16(&W[(size_t)(kc + bRow[j]) * NDIM + nBase + bCol[j]],
                         &sB[buf][bRow[j] * NT + bCol[j]]);
#else
        #pragma unroll
        for (int j = 0; j < 2; ++j)
            *reinterpret_cast<float4*>(&sA[buf][aRow[j] * KC + aCol[j]]) =
                *reinterpret_cast<const float4*>(&A[(size_t)aRow[j] * KDIM + kc + aCol[j]]);
        #pragma unroll
        for (int j = 0; j < 4; ++j)
            *reinterpret_cast<float4*>(&sB[buf][bRow[j] * NT + bCol[j]]) =
                *reinterpret_cast<const float4*>(&W[(size_t)(kc + bRow[j]) * NDIM + nBase + bCol[j]]);
#endif
    };

    v8f acc[4] = {};

    stage(0, 0);
    async_wait_all();
    __syncthreads();

    int cur = 0;
    for (int kc = 0; kc < KDIM; kc += KC) {
        if (kc + KC < KDIM) stage(kc + KC, cur ^ 1);   // prefetch next chunk

        #pragma unroll
        for (int k = 0; k < KC; k += 4) {
            v2f bf;
            bf.x = sB[cur][(k + koff)     * NT + nLoc];
            bf.y = sB[cur][(k + koff + 1) * NT + nLoc];
            #pragma unroll
            for (int mt = 0; mt < 4; ++mt) {
                int m = mt * 16 + (lane & 15);
                v2f af;
                af.x = sA[cur][m * KC + k + koff];
                af.y = sA[cur][m * KC + k + koff + 1];
                acc[mt] = __builtin_amdgcn_wmma_f32_16x16x4_f32(
                    false, af, false, bf, (short)0, acc[mt], false, false);
            }
        }

        async_wait_all();      // next chunk's tiles landed in LDS
        __syncthreads();       // everyone done reading cur, data visible
        cur ^= 1;
    }

    const int n  = nBase + nLoc;
    const float bn = bias[n];
    #pragma unroll
    for (int mt = 0; mt < 4; ++mt) {
        #pragma unroll
        for (int v = 0; v < 8; ++v) {
            int row = mt * 16 + v + 8 * (lane >> 4);   // C/D layout: VGPR v -> M=v / v+8
            C[(size_t)row * NDIM + n] = acc[mt][v] + bn;
        }
    }
}

// =====================================================================
// Kernel C: L2-normalize couplings rows, scatter into dense K (with
// duplicate-index accumulation), count nonzeros per row.
// Each thread owns one (b,i) row of K -> no atomics needed.
// =====================================================================
__global__ __launch_bounds__(256)
void scatter_kernel(const float* __restrict__ coupl, // [64, 8192] = [B,N,32]
                    const int*   __restrict__ conn,  // [256, 32]
                    float* __restrict__ Kout,        // [64, 256, 256]
                    float* __restrict__ nnz) {       // [64, 256]
    const int b = blockIdx.x;
    const int i = threadIdx.x;

    const float* c = &coupl[(size_t)b * FEAT + i * NUM_CN];
    float v[NUM_CN];
    float ss = 0.f;
    #pragma unroll
    for (int j = 0; j < NUM_CN; ++j) { v[j] = c[j]; ss += v[j] * v[j]; }
    const float inv = rsqrtf(ss);

    float* Krow = &Kout[((size_t)b * N_OSC + i) * N_OSC];
    const float4 z = make_float4(0.f, 0.f, 0.f, 0.f);
    #pragma unroll 8
    for (int m = 0; m < N_OSC; m += 4)
        *reinterpret_cast<float4*>(&Krow[m]) = z;

    #pragma unroll
    for (int j = 0; j < NUM_CN; ++j)
        Krow[conn[i * NUM_CN + j]] += v[j] * inv;

    int cnt = 0;
    for (int m = 0; m < N_OSC; ++m) cnt += (Krow[m] != 0.f) ? 1 : 0;
    nnz[b * N_OSC + i] = (float)cnt;
}

// =====================================================================
// Kernel D: 40-step Euler integration of Kuramoto dynamics.
// K[b] = 256KB lives entirely in CDNA5's 320KB LDS (rows padded to 260
// floats to break the stride-256 bank conflict). One WG per batch; the
// one-time K load uses the async global->LDS path.
// =====================================================================
#define KROW 260
__global__ __launch_bounds__(256)
void kuramoto_kernel(const float* __restrict__ Kmat,   // [64,256,256]
                     const float* __restrict__ nnz,    // [64,256]
                     const float* __restrict__ phase0, // [64,256]
                     float* __restrict__ phases) {     // [40,64,256]
    extern __shared__ float smem[];
    float* sK   = smem;                    // 256 * KROW
    float* sSin = smem + N_OSC * KROW;     // 256
    float* sCos = sSin + N_OSC;            // 256

    const int b = blockIdx.x;
    const int i = threadIdx.x;

    const float* g = &Kmat[((size_t)b * N_OSC + i) * N_OSC];
#if HAVE_ASYNC_LDS
    for (int m = 0; m < N_OSC; m += 4)
        async_copy16(&g[m], &sK[i * KROW + m]);
    async_wait_all();
#else
    #pragma unroll 4
    for (int m = 0; m < N_OSC; m += 4)
        *reinterpret_cast<float4*>(&sK[i * KROW + m]) =
            *reinterpret_cast<const float4*>(&g[m]);
#endif

    float phase = phase0[b * N_OSC + i];
    const float inv_n = 1.0f / nnz[b * N_OSC + i];
    __syncthreads();

    for (int ts = 0; ts < T_STEPS; ++ts) {
        const float si = sinf(phase);
        const float ci = cosf(phase);
        sSin[i] = si;
        sCos[i] = ci;
        __syncthreads();

        float sum_s = 0.f, sum_c = 0.f;
        #pragma unroll 4
        for (int m = 0; m < N_OSC; m += 4) {
            float4 kk = *reinterpret_cast<float4*>(&sK[i * KROW + m]);
            float4 sv = *reinterpret_cast<float4*>(&sSin[m]);
            float4 cv = *reinterpret_cast<float4*>(&sCos[m]);
            sum_s += kk.x * sv.x + kk.y * sv.y + kk.z * sv.z + kk.w * sv.w;
            sum_c += kk.x * cv.x + kk.y * cv.y + kk.z * cv.z + kk.w * cv.w;
        }
        const float dphi = (sum_s * ci - sum_c * si) * inv_n;
        phase += DT_STEP * dphi;
        phases[((size_t)ts * B_SZ + b) * N_OSC + i] = phase;
        __syncthreads();
    }
}

// =====================================================================
extern "C" void kernel_launch(void* const* d_in, const int* in_sizes, int n_in,
                              void* d_out, int out_size, void* d_ws, size_t ws_size,
                              hipStream_t stream) {
    const float* x       = (const float*)d_in[0];
    const float* conv_w0 = (const float*)d_in[1];
    const float* conv_b0 = (const float*)d_in[2];
    const float* conv_w1 = (const float*)d_in[3];
    const float* conv_b1 = (const float*)d_in[4];
    const float* lin_w   = (const float*)d_in[5];
    const float* lin_b   = (const float*)d_in[6];
    const int*   conn    = (const int*)d_in[7];
    const float* phase0  = (const float*)d_in[8];

    float* feat  = (float*)d_ws;                          // 64*8192
    float* coupl = feat  + (size_t)B_SZ * FEAT;           // 64*8192
    float* nnz   = coupl + (size_t)B_SZ * FEAT;           // 64*256

    float* phases = (float*)d_out;                        // 40*64*256
    float* Kout   = phases + (size_t)T_STEPS * B_SZ * N_OSC; // 64*256*256

    conv_fused_kernel<<<B_SZ, 256, 0, stream>>>(x, conv_w0, conv_b0,
                                                conv_w1, conv_b1, feat);
    gemm_wmma_kernel<<<NDIM / NT, 256, 0, stream>>>(feat, lin_w, lin_b, coupl);
    scatter_kernel<<<B_SZ, 256, 0, stream>>>(coupl, conn, Kout, nnz);

    const size_t lds = (size_t)(N_OSC * KROW + 2 * N_OSC) * sizeof(float);
    kuramoto_kernel<<<B_SZ, 256, lds, stream>>>(Kout, nnz, phase0, phases);
}